// Bicubic_Interpolator_76020921139567
// MI455X (gfx1250) — compile-verified
//
#include <hip/hip_runtime.h>
#include <stdint.h>

// Problem constants from the reference: B=8, C=3, H=W=1024.
#define Bn 8
#define Cn 3
#define Hn 1024
#define Wn 1024
#define TILE_W 32   // one wave per tile row (wave32)
#define TILE_H 8    // 256 threads / block = 8 waves

typedef unsigned int uint32x4 __attribute__((ext_vector_type(4)));
typedef int          int32x8  __attribute__((ext_vector_type(8)));
typedef int          int32x4  __attribute__((ext_vector_type(4)));

// Low 32 bits of a generic (flat) shared pointer == LDS byte offset
// (LDS aperture occupies the high 32 bits on CDNA5).
__device__ __forceinline__ uint32_t lds_byte_addr(const void* p) {
    return (uint32_t)(uintptr_t)p;
}

// Build a 2D-tile Tensor-DMA descriptor (D#) and issue TENSOR_LOAD_TO_LDS.
//   gsrc      : global address of the tile's top-left element
//   ldsAddr   : LDS byte offset of destination (tile stored row-major, contiguous)
//   remW/remH : elements remaining in the tensor from the tile origin (OOB bounds)
__device__ __forceinline__ void tdm_load_tile_f32(const float* gsrc,
                                                  uint32_t ldsAddr,
                                                  uint32_t remW, uint32_t remH) {
    const uint64_t ga      = (uint64_t)(uintptr_t)gsrc;
    const uint64_t stride1 = (uint64_t)Hn * (uint64_t)Wn;   // dim1 stride (elements)

    // ---- D# group 0 (128b): count / lds_addr / global_addr / type=2 ----
    uint32x4 g0;
    g0[0] = 1u;                                             // count=1 (valid, user mode)
    g0[1] = ldsAddr;                                        // lds_addr (bytes)
    g0[2] = (uint32_t)(ga & 0xFFFFFFFFull);                 // global_addr[31:0]
    g0[3] = (uint32_t)((ga >> 32) & 0x01FFFFFFull)          // global_addr[56:32]
          | (2u << 30);                                     // type = 2 ("image")

    // ---- D# group 1 (256b): data_size / dims / tile dims / strides ----
    int32x8 g1;
    g1[0] = (int)(2u << 16);                                // data_size=2 (4 bytes); wg_mask=0
    g1[1] = (int)((remW & 0xFFFFu) << 16);                  // tensor_dim0[15:0] @ bits 63:48
    g1[2] = (int)(((remW >> 16) & 0xFFFFu)                  // tensor_dim0[31:16]
          | ((remH & 0xFFFFu) << 16));                      // tensor_dim1[15:0]
    g1[3] = (int)(((remH >> 16) & 0xFFFFu)                  // tensor_dim1[31:16]
          | ((uint32_t)TILE_W << 16));                      // tile_dim0
    g1[4] = (int)TILE_H;                                    // tile_dim1 ; tile_dim2 = 0
    g1[5] = (int)Wn;                                        // tensor_dim0_stride[31:0]
    g1[6] = (int)(((uint32_t)(stride1 & 0xFFFFull)) << 16); // stride0[47:32]=0 ; stride1[15:0]
    g1[7] = (int)(uint32_t)(stride1 >> 16);                 // tensor_dim1_stride[47:16]

    const int32x4 gz4 = {0, 0, 0, 0};                       // groups 2/3 unused (2D tile)
    const int32x8 gz8 = {0, 0, 0, 0, 0, 0, 0, 0};
    __builtin_amdgcn_tensor_load_to_lds(g0, g1, gz4, gz4, gz8, 0);
}

__device__ __forceinline__ void cubic_weights(float t, float w[4]) {
    const float t2 = t * t;
    const float t3 = t2 * t;
    w[0] = (-t3 + 2.0f * t2 - t) * 0.5f;
    w[1] = (3.0f * t3 - 5.0f * t2 + 2.0f) * 0.5f;
    w[2] = (-3.0f * t3 + 4.0f * t2 + t) * 0.5f;
    w[3] = 1.0f - (w[0] + w[1] + w[2]);
}

__global__ __launch_bounds__(TILE_W * TILE_H)
void bicubic_warp_kernel(const float* __restrict__ img,
                         const float* __restrict__ dxp,
                         const float* __restrict__ dyp,
                         float* __restrict__ out) {
    __shared__ float sdx[TILE_H][TILE_W];
    __shared__ float sdy[TILE_H][TILE_W];

    const int plane = blockIdx.z;                 // b*C + c
    const int x0t   = blockIdx.x * TILE_W;
    const int y0t   = blockIdx.y * TILE_H;
    const size_t planeOff = (size_t)plane * (size_t)Hn * (size_t)Wn;
    const size_t tileOff  = planeOff + (size_t)y0t * Wn + (size_t)x0t;

    // Wave 0 (threadIdx.y == 0 is wave-uniform: blockDim.x == 32) issues the
    // Tensor-DMA loads of this block's delta tiles into LDS, then drains its
    // TENSORcnt before releasing the workgroup barrier.
    if (threadIdx.y == 0) {
        tdm_load_tile_f32(dxp + tileOff, lds_byte_addr(&sdx[0][0]),
                          (uint32_t)(Wn - x0t), (uint32_t)(Hn - y0t));
        tdm_load_tile_f32(dyp + tileOff, lds_byte_addr(&sdy[0][0]),
                          (uint32_t)(Wn - x0t), (uint32_t)(Hn - y0t));
        __builtin_amdgcn_s_wait_tensorcnt(0);
    }
    __syncthreads();

    const int tx = threadIdx.x;
    const int ty = threadIdx.y;
    const int x  = x0t + tx;
    const int y  = y0t + ty;

    const float dx = sdx[ty][tx];
    const float dy = sdy[ty][tx];

    // Coordinate mapping — same arithmetic as the reference.
    const float nx = ((float)x + dx - (float)(Wn / 2)) / ((float)(Wn / 2) - 1.0f);
    const float ny = ((float)y + dy - (float)(Hn / 2)) / ((float)(Hn / 2) - 1.0f);
    const float xm = (nx + 1.0f) * ((float)(Wn - 1) * 0.5f);
    const float ym = (ny + 1.0f) * ((float)(Hn - 1) * 0.5f);

    const float xf = floorf(xm);
    const float yf = floorf(ym);
    const float ftx = xm - xf;
    const float fty = ym - yf;
    const int xi = (int)xf;
    const int yi = (int)yf;

    float wx[4], wy[4];
    cubic_weights(ftx, wx);
    cubic_weights(fty, wy);

    int xs[4], ys[4];
#pragma unroll
    for (int i = 0; i < 4; ++i) {
        xs[i] = min(max(xi + i - 1, 0), Wn - 1);
        ys[i] = min(max(yi + i - 1, 0), Hn - 1);
    }

    // 16 clamped taps. Default (RT) temporal policy: the 24 image planes total
    // 96 MB and live in the 192 MB L2, so gathers are L0/L2 hits.
    const float* __restrict__ base = img + planeOff;
    float acc = 0.0f;
#pragma unroll
    for (int j = 0; j < 4; ++j) {
        const float* __restrict__ rp = base + (size_t)ys[j] * Wn;
        float row = wx[0] * rp[xs[0]];
        row = fmaf(wx[1], rp[xs[1]], row);
        row = fmaf(wx[2], rp[xs[2]], row);
        row = fmaf(wx[3], rp[xs[3]], row);
        acc = fmaf(wy[j], row, acc);
    }
    acc = fminf(fmaxf(acc, 0.0f), 1.0f);

    // Streamed output: non-temporal so it doesn't evict the image planes from L2.
    __builtin_nontemporal_store(acc, out + planeOff + (size_t)y * Wn + (size_t)x);
}

extern "C" void kernel_launch(void* const* d_in, const int* in_sizes, int n_in,
                              void* d_out, int out_size, void* d_ws, size_t ws_size,
                              hipStream_t stream) {
    (void)in_sizes; (void)n_in; (void)out_size; (void)d_ws; (void)ws_size;
    const float* img = (const float*)d_in[0];   // input_image [8,3,1024,1024] f32
    const float* dx  = (const float*)d_in[1];   // delta_x     [8,3,1024,1024] f32
    const float* dy  = (const float*)d_in[2];   // delta_y     [8,3,1024,1024] f32
    float* outp      = (float*)d_out;

    dim3 block(TILE_W, TILE_H, 1);
    dim3 grid(Wn / TILE_W, Hn / TILE_H, Bn * Cn);
    bicubic_warp_kernel<<<grid, block, 0, stream>>>(img, dx, dy, outp);
}